// Decoder_85968065397056
// MI455X (gfx1250) — compile-verified
//
#include <hip/hip_runtime.h>
#include <hip/hip_bf16.h>

#define MAT_N 1024
#define NMAT  16
#define PB    32     // LU panel width (multiple of 4 for K=4 f32 WMMA)

typedef float v2f __attribute__((ext_vector_type(2)));
typedef float v8f __attribute__((ext_vector_type(8)));

// ---------------------------------------------------------------------------
// Kernel 1: blocked LU (partial pivoting) -> log|det|.
// One 1024-thread block per matrix. Sign is irrelevant (reference discards it).
// Panel (width 32) factored classically; the rank-32 trailing update runs on
// V_WMMA_F32_16X16X4_F32. Factored columns store -L so the WMMA update is
// C += (-L) @ U with no extra VALU, and trsm uses u[r] += sL[r][t]*u[t].
// ---------------------------------------------------------------------------
__global__ __launch_bounds__(1024) void lu_logdet_kernel(
    const float* __restrict__ x, float* __restrict__ work, float* __restrict__ slogdet_out)
{
    const int m    = blockIdx.x;
    const int tid  = threadIdx.x;
    const int lane = tid & 31;
    const int wid  = tid >> 5;
    float* A = work + (size_t)m * MAT_N * MAT_N;
    const float* X = x + (size_t)m * MAT_N * MAT_N;

    for (int i = tid; i < MAT_N * MAT_N; i += 1024) A[i] = X[i];
    __syncthreads();

    __shared__ float s_red_v[32];
    __shared__ int   s_red_i[32];
    __shared__ int   s_pivrow;
    __shared__ float s_prow[PB];          // pivot row, panel columns
    __shared__ float s_L[PB][PB + 1];     // -L11 (unit diag implied), padded

    float logabs = 0.0f;

    for (int k0 = 0; k0 < MAT_N; k0 += PB) {
        const int k1 = k0 + PB;

        // ================= panel factorization: columns k0..k1-1 ==========
        for (int kk = 0; kk < PB; ++kk) {
            const int k = k0 + kk;

            // ---- pivot search: argmax |A[i][k]|, i = k..N-1
            int   ci = k + tid;
            float cv = (ci < MAT_N) ? fabsf(A[(size_t)ci * MAT_N + k]) : -1.0f;
            for (int off = 16; off > 0; off >>= 1) {
                float ov = __shfl_xor(cv, off, 32);
                int   oi = __shfl_xor(ci, off, 32);
                if (ov > cv || (ov == cv && oi < ci)) { cv = ov; ci = oi; }
            }
            if (lane == 0) { s_red_v[wid] = cv; s_red_i[wid] = ci; }
            __syncthreads();
            if (wid == 0) {
                cv = s_red_v[lane]; ci = s_red_i[lane];
                for (int off = 16; off > 0; off >>= 1) {
                    float ov = __shfl_xor(cv, off, 32);
                    int   oi = __shfl_xor(ci, off, 32);
                    if (ov > cv || (ov == cv && oi < ci)) { cv = ov; ci = oi; }
                }
                if (lane == 0) s_pivrow = ci;
            }
            __syncthreads();
            const int piv = s_pivrow;

            // ---- swap rows k<->piv over cols >= k0 (panel L must follow
            //      pivoting; cols < k0 never re-read). Stage panel pivot row.
            for (int j = k0 + tid; j < MAT_N; j += 1024) {
                float akj = A[(size_t)k * MAT_N + j];
                if (piv != k) {
                    float apj = A[(size_t)piv * MAT_N + j];
                    A[(size_t)k   * MAT_N + j] = apj;
                    A[(size_t)piv * MAT_N + j] = akj;
                    akj = apj;
                }
                if (j < k1) s_prow[j - k0] = akj;
            }
            __syncthreads();

            const float pv = s_prow[kk];
            logabs += __logf(fabsf(pv));              // identical in all threads
            const float inv = (pv != 0.0f) ? (1.0f / pv) : 0.0f;

            // ---- rank-1 update confined to panel; store -L in column k
            for (int i = k + 1 + wid; i < MAT_N; i += 32) {
                const size_t rb = (size_t)i * MAT_N;
                const float l = A[rb + k] * inv;      // broadcast read per wave
                if (lane == 0) A[rb + k] = -l;        // store -L
                if (lane > kk) A[rb + k0 + lane] -= l * s_prow[lane];
            }
            __syncthreads();
        }

        // ================= stage -L11; trsm U12 = L11^-1 * A[k0:k1, k1:] ===
        {
            const int r = tid >> 5, c = tid & 31;
            s_L[r][c] = (c < r) ? A[(size_t)(k0 + r) * MAT_N + k0 + c] : 0.0f;
        }
        __syncthreads();

        for (int j = k1 + tid; j < MAT_N; j += 1024) {
            float u[PB];
            #pragma unroll
            for (int t = 0; t < PB; ++t) u[t] = A[(size_t)(k0 + t) * MAT_N + j];
            #pragma unroll
            for (int r = 1; r < PB; ++r) {
                float acc = u[r];
                #pragma unroll
                for (int t = 0; t < r; ++t) acc += s_L[r][t] * u[t]; // s_L = -L
                u[r] = acc;
                A[(size_t)(k0 + r) * MAT_N + j] = acc;
            }
        }
        __syncthreads();

        // ================= trailing update: C += (-L21) @ U12 (WMMA) ======
        const int T = MAT_N - k1;                 // multiple of 32 (or 0)
        if (T > 0) {
            const int nsd = T >> 5;               // super-blocks per dim
            const int nsb = nsd * nsd;
            const int lm    = lane & 15;
            const int koff  = (lane >> 4) << 1;   // K sub-offset 0 or 2
            const int rhalf = (lane >> 4) << 3;   // C/D row offset 0 or 8

            for (int sb = wid; sb < nsb; sb += 32) {      // wave-uniform
                const int sbr = k1 + (sb / nsd) * 32;
                const int sbc = k1 + (sb % nsd) * 32;

                v8f acc2[2][2];
                #pragma unroll
                for (int mi = 0; mi < 2; ++mi)
                    #pragma unroll
                    for (int ni = 0; ni < 2; ++ni)
                        #pragma unroll
                        for (int r = 0; r < 8; ++r)
                            acc2[mi][ni][r] =
                                A[(size_t)(sbr + mi * 16 + r + rhalf) * MAT_N
                                  + sbc + ni * 16 + lm];

                #pragma unroll
                for (int kk2 = 0; kk2 < PB; kk2 += 4) {
                    v2f af[2], bf[2];
                    #pragma unroll
                    for (int mi = 0; mi < 2; ++mi) {
                        const float* ap = A + (size_t)(sbr + mi * 16 + lm) * MAT_N
                                            + k0 + kk2 + koff;
                        af[mi] = *(const v2f*)ap;          // -L21, 8B aligned
                    }
                    #pragma unroll
                    for (int ni = 0; ni < 2; ++ni) {
                        const float* bp = A + (size_t)(k0 + kk2 + koff) * MAT_N
                                            + sbc + ni * 16 + lm;
                        bf[ni].x = bp[0];
                        bf[ni].y = bp[MAT_N];
                    }
                    #pragma unroll
                    for (int mi = 0; mi < 2; ++mi)
                        #pragma unroll
                        for (int ni = 0; ni < 2; ++ni)
                            acc2[mi][ni] = __builtin_amdgcn_wmma_f32_16x16x4_f32(
                                false, af[mi], false, bf[ni], (short)0,
                                acc2[mi][ni], false, false);
                }

                #pragma unroll
                for (int mi = 0; mi < 2; ++mi)
                    #pragma unroll
                    for (int ni = 0; ni < 2; ++ni)
                        #pragma unroll
                        for (int r = 0; r < 8; ++r)
                            A[(size_t)(sbr + mi * 16 + r + rhalf) * MAT_N
                              + sbc + ni * 16 + lm] = acc2[mi][ni][r];
            }
        }
        __syncthreads();
    }

    if (tid == 0) slogdet_out[m] = logabs;
}

// ---------------------------------------------------------------------------
// Kernel 2: gate + top-8 selection + fold MLP weights (weff = W2@W1, beff).
// ---------------------------------------------------------------------------
__global__ void select_kernel(
    const unsigned char* __restrict__ flags,
    const float* __restrict__ W1, const float* __restrict__ b1,
    const float* __restrict__ W2, const float* __restrict__ b2,
    const float* __restrict__ slogdet,
    float* __restrict__ meta_f, int* __restrict__ meta_i)
{
    float scores[NMAT];
    int count = 0;
    for (int i = 0; i < NMAT; ++i) {
        const bool act = flags[i] != 0;
        count += act ? 1 : 0;
        scores[i] = act ? slogdet[i] : -__builtin_inff();
    }
    meta_i[8] = (count >= 4) ? 1 : 0;

    bool used[NMAT];
    for (int i = 0; i < NMAT; ++i) used[i] = false;
    for (int j = 0; j < 8; ++j) {             // lax.top_k: ties -> lowest index
        int best = -1; float bv = 0.0f;
        for (int i = 0; i < NMAT; ++i) {
            if (used[i]) continue;
            if (best < 0 || scores[i] > bv) { best = i; bv = scores[i]; }
        }
        used[best] = true;
        meta_i[j] = best;
    }

    for (int c = 0; c < 10; ++c) {            // weff[c] = sum_o W2[o]*W1[o][c]
        float s = 0.0f;
        for (int o = 0; o < 16; ++o) s += W2[o] * W1[o * 10 + c];
        meta_f[16 + c] = s;
    }
    float be = b2[0];
    for (int o = 0; o < 16; ++o) be += W2[o] * b1[o];
    meta_f[26] = be;
}

// ---------------------------------------------------------------------------
// Kernel 3: fused  out = swish( sum_p weff[p]*(A_p@B_p) + sum_q weff[6+q]*P_q + beff )
// f32 WMMA, wave tile 64x64 (4x4 frags), 256 waves, data L2-resident (32MB).
// ---------------------------------------------------------------------------
__global__ __launch_bounds__(256) void fused_gemm_kernel(
    const float* __restrict__ x, float* __restrict__ out,
    const float* __restrict__ meta_f, const int* __restrict__ meta_i)
{
    const int lane = threadIdx.x & 31;
    const int wave = blockIdx.x * (blockDim.x >> 5) + (threadIdx.x >> 5); // 0..255
    const int wm = wave >> 4;
    const int wn = wave & 15;
    const int row0 = wm * 64;
    const int col0 = wn * 64;
    const int lm    = lane & 15;
    const int koff  = (lane >> 4) << 1;
    const int rhalf = (lane >> 4) << 3;

    const int gate = meta_i[8];
    if (!gate) {                          // inactive: zero output, wave-uniform
        for (int mi = 0; mi < 4; ++mi)
            for (int r = 0; r < 8; ++r) {
                const size_t rb = (size_t)(row0 + mi * 16 + r + rhalf) * MAT_N;
                for (int ni = 0; ni < 4; ++ni)
                    out[rb + col0 + ni * 16 + lm] = 0.0f;
            }
        return;
    }

    v8f acc[4][4];
    #pragma unroll
    for (int mi = 0; mi < 4; ++mi)
        #pragma unroll
        for (int ni = 0; ni < 4; ++ni)
            acc[mi][ni] = (v8f){0.f, 0.f, 0.f, 0.f, 0.f, 0.f, 0.f, 0.f};

    const int p1[6] = {0, 0, 0, 1, 1, 2};
    const int p2[6] = {1, 2, 3, 2, 3, 3};

    for (int p = 0; p < 6; ++p) {
        const float* Amat = x + (size_t)meta_i[p1[p]] * MAT_N * MAT_N;
        const float* Bmat = x + (size_t)meta_i[p2[p]] * MAT_N * MAT_N;
        const float s = meta_f[16 + p];       // fold weff[p] into A

        for (int k0 = 0; k0 < MAT_N; k0 += 4) {
            v2f a[4], b[4];
            #pragma unroll
            for (int mi = 0; mi < 4; ++mi) {
                const float* ap = Amat + (size_t)(row0 + mi * 16 + lm) * MAT_N + k0 + koff;
                v2f t = *(const v2f*)ap;      // 8B aligned
                a[mi].x = t.x * s;
                a[mi].y = t.y * s;
            }
            #pragma unroll
            for (int ni = 0; ni < 4; ++ni) {
                const float* bp = Bmat + (size_t)(k0 + koff) * MAT_N + col0 + ni * 16 + lm;
                b[ni].x = bp[0];
                b[ni].y = bp[MAT_N];
            }
            #pragma unroll
            for (int mi = 0; mi < 4; ++mi)
                #pragma unroll
                for (int ni = 0; ni < 4; ++ni)
                    acc[mi][ni] = __builtin_amdgcn_wmma_f32_16x16x4_f32(
                        false, a[mi], false, b[ni], (short)0, acc[mi][ni],
                        false, false);
        }
    }

    // ---- epilogue: preserve channels + bias + swish
    const float w6 = meta_f[22], w7 = meta_f[23], w8 = meta_f[24], w9 = meta_f[25];
    const float beff = meta_f[26];
    const float* P0 = x + (size_t)meta_i[4] * MAT_N * MAT_N;
    const float* P1 = x + (size_t)meta_i[5] * MAT_N * MAT_N;
    const float* P2 = x + (size_t)meta_i[6] * MAT_N * MAT_N;
    const float* P3 = x + (size_t)meta_i[7] * MAT_N * MAT_N;

    for (int mi = 0; mi < 4; ++mi) {
        for (int r = 0; r < 8; ++r) {
            const size_t rb = (size_t)(row0 + mi * 16 + r + rhalf) * MAT_N;
            for (int ni = 0; ni < 4; ++ni) {
                const size_t idx = rb + col0 + ni * 16 + lm;
                float v = acc[mi][ni][r] + beff
                        + w6 * P0[idx] + w7 * P1[idx]
                        + w8 * P2[idx] + w9 * P3[idx];
                out[idx] = v / (1.0f + __expf(-v));    // swish
            }
        }
    }
}

// ---------------------------------------------------------------------------
// Kernel 4: out_active flag (tuple element 2) appended after the N*N image.
// ---------------------------------------------------------------------------
__global__ void write_flag_kernel(float* __restrict__ out, const int* __restrict__ meta_i)
{
    out[(size_t)MAT_N * MAT_N] = meta_i[8] ? 1.0f : 0.0f;
}

// ---------------------------------------------------------------------------
extern "C" void kernel_launch(void* const* d_in, const int* in_sizes, int n_in,
                              void* d_out, int out_size, void* d_ws, size_t ws_size,
                              hipStream_t stream)
{
    const float*         x     = (const float*)d_in[0];
    const unsigned char* flags = (const unsigned char*)d_in[1]; // jnp bool = 1 byte
    const float*         W1    = (const float*)d_in[2];
    const float*         b1    = (const float*)d_in[3];
    const float*         W2    = (const float*)d_in[4];
    const float*         b2    = (const float*)d_in[5];
    float* out = (float*)d_out;

    // workspace: [16 * N*N floats LU scratch][meta_f: 32 floats][meta_i: 16 ints]
    float* ws_f   = (float*)d_ws;
    float* meta_f = ws_f + (size_t)NMAT * MAT_N * MAT_N; // [0..15] slogdet, [16..25] weff, [26] beff
    int*   meta_i = (int*)(meta_f + 32);                 // [0..7] top-8 idx, [8] gate

    lu_logdet_kernel<<<NMAT, 1024, 0, stream>>>(x, ws_f, meta_f);
    select_kernel<<<1, 1, 0, stream>>>(flags, W1, b1, W2, b2, meta_f, meta_f, meta_i);
    fused_gemm_kernel<<<32, 256, 0, stream>>>(x, out, meta_f, meta_i);
    if (out_size > MAT_N * MAT_N)
        write_flag_kernel<<<1, 1, 0, stream>>>(out, meta_i);

    (void)in_sizes; (void)n_in; (void)ws_size;
}